// MAA_layer_75522704933057
// MI455X (gfx1250) — compile-verified
//
#include <hip/hip_runtime.h>

// MAA layer, factorized:
//   q_{t+1} = A_t q_t,  A_t = p_t*S + (1-p_t)*I          (forward scan)
//   v_{t-1} = A_t^T v_t, v_{T-1}[i] = 1/i                (backward scan, independent!)
//   c_t = p_t * sum_j q_t[j] * v_t[j+1]                  (parallel dots)
//   out  = X^T c                                         (f32 WMMA matvec)
//
// T = 1024, D = 256, L = T+1. q_t support is i<=t<=1023 so both scans track
// exactly 1024 f32 values = 32 lanes x 32 registers of ONE wave32; the shift
// needs only a single __shfl per step (no LDS, no barriers).

#define CHUNK 32   // f32 elements per lane; requires T == 32*CHUNK == 1024

__global__ __launch_bounds__(32)
void maa_scan_kernel(const float* __restrict__ p,
                     float* __restrict__ Q,   // [T][T] rows: q_t[0..T-1]
                     float* __restrict__ V,   // [T][T] rows: v_t[1..T]
                     int T) {
  const int lane = threadIdx.x;

  if (blockIdx.x == 0) {
    // ---- forward scan: q_0 = delta_0; store q_t, then q <- A_{p[t]} q ----
    float q[CHUNK];
    #pragma unroll
    for (int i = 0; i < CHUNK; ++i) q[i] = 0.f;
    if (lane == 0) q[0] = 1.f;

    for (int t = 0; t < T; ++t) {
      float4* row = (float4*)(Q + (size_t)t * T + lane * CHUNK);
      #pragma unroll
      for (int j = 0; j < CHUNK / 4; ++j)
        row[j] = make_float4(q[4*j], q[4*j+1], q[4*j+2], q[4*j+3]);

      const float pt = p[t];
      const float om = 1.f - pt;
      float bnd = __shfl_up(q[CHUNK-1], 1);      // q[32*lane - 1] from lane-1
      if (lane == 0) bnd = 0.f;                  // q[-1] = 0
      #pragma unroll
      for (int i = CHUNK - 1; i >= 1; --i)       // descending: q[i-1] still old
        q[i] = om * q[i] + pt * q[i-1];
      q[0] = om * q[0] + pt * bnd;
    }
    asm volatile("s_wait_storecnt 0" ::: "memory");
  } else {
    // ---- backward scan: v[g], g = 32*lane+1+i in [1..1024]; v[1025]=0 ----
    float v[CHUNK];
    #pragma unroll
    for (int i = 0; i < CHUNK; ++i)
      v[i] = 1.f / (float)(lane * CHUNK + 1 + i);   // h[g] = 1/g

    for (int t = T - 1; t >= 0; --t) {
      float4* row = (float4*)(V + (size_t)t * T + lane * CHUNK);
      #pragma unroll
      for (int j = 0; j < CHUNK / 4; ++j)
        row[j] = make_float4(v[4*j], v[4*j+1], v[4*j+2], v[4*j+3]);

      const float pt = p[t];
      const float om = 1.f - pt;
      float bnd = __shfl_down(v[0], 1);          // v[32*lane + 33] from lane+1
      if (lane == 31) bnd = 0.f;                 // v[1025] = 0
      #pragma unroll
      for (int i = 0; i < CHUNK - 1; ++i)        // ascending: v[i+1] still old
        v[i] = om * v[i] + pt * v[i+1];
      v[CHUNK-1] = om * v[CHUNK-1] + pt * bnd;
    }
    asm volatile("s_wait_storecnt 0" ::: "memory");
  }
}

// c_t = p_t * dot(Q[t], V[t])  (V row already holds v_t shifted by +1).
// One wave per t; rows are L2-resident (8 MB << 192 MB).
__global__ __launch_bounds__(256)
void maa_dots_kernel(const float* __restrict__ Q,
                     const float* __restrict__ V,
                     const float* __restrict__ p,
                     float* __restrict__ c, int T) {
  const int t = (blockIdx.x * blockDim.x + threadIdx.x) >> 5;
  const int lane = threadIdx.x & 31;
  if (t >= T) return;
  const float4* qr = (const float4*)(Q + (size_t)t * T + lane * CHUNK);
  const float4* vr = (const float4*)(V + (size_t)t * T + lane * CHUNK);
  float partial = 0.f;
  #pragma unroll
  for (int j = 0; j < CHUNK / 4; ++j) {
    float4 a = qr[j], b = vr[j];
    partial += a.x*b.x + a.y*b.y + a.z*b.z + a.w*b.w;
  }
  #pragma unroll
  for (int off = 16; off >= 1; off >>= 1)
    partial += __shfl_xor(partial, off);
  if (lane == 0) c[t] = p[t] * partial;
}

// out = X^T c via V_WMMA_F32_16X16X4_F32 (full f32, matches reference precision).
// A (16x4): A[m][k] = x[t0+k][d0+m]; per ISA layout lane L holds M=L%16,
// VGPR0/1 = K{0,1} for L<16, K{2,3} for L>=16. B (4x16): c replicated over N,
// so every column of C equals the output slice; lanes with N==0 write it.
//
// Addressing: all offsets fit in 32 bits (x is 1 MB), so march a 32-bit
// per-lane offset by a constant each iteration -> GVS-mode global loads
// (SGPR base + 32-bit VGPR offset), no per-iteration 64-bit VALU math.
typedef float v2f __attribute__((ext_vector_type(2)));
typedef float v8f __attribute__((ext_vector_type(8)));

__global__ __launch_bounds__(32)
void maa_matvec_wmma(const float* __restrict__ x,
                     const float* __restrict__ c,
                     float* __restrict__ out,
                     int T, int D) {
  const int lane = threadIdx.x;
  const int d0 = blockIdx.x * 16;
  const int m  = lane & 15;
  const int kb = (lane < 16) ? 0 : 2;      // K-pair owned by this half-wave

  const float* xbase = x + d0 + m;         // per-lane column base
  unsigned xoff = (unsigned)(kb * D);      // marches by 4*D per WMMA step

  v8f acc = {};
  #pragma unroll 4
  for (int t0 = 0; t0 < T; t0 += 4) {
    v2f A, B;
    A[0] = xbase[xoff];                    // x[t0+kb  ][d0+m]
    A[1] = xbase[xoff + (unsigned)D];      // x[t0+kb+1][d0+m]
    B = *(const v2f*)(c + t0 + kb);        // 8B-aligned: c[t0+kb], c[t0+kb+1]
    // 8 args: (neg_a, A, neg_b, B, c_mod, C, reuse_a, reuse_b)
    acc = __builtin_amdgcn_wmma_f32_16x16x4_f32(
        false, A, false, B, (short)0, acc, false, false);
    xoff += 4u * (unsigned)D;
  }

  // C layout: lane L, VGPR r -> M = r + (L<16 ? 0 : 8), N = L%16. Take N==0.
  if ((lane & 15) == 0) {
    const int mb = (lane < 16) ? 0 : 8;
    float4* o = (float4*)(out + d0 + mb);  // 32B-aligned
    o[0] = make_float4(acc[0], acc[1], acc[2], acc[3]);
    o[1] = make_float4(acc[4], acc[5], acc[6], acc[7]);
  }
}

extern "C" void kernel_launch(void* const* d_in, const int* in_sizes, int n_in,
                              void* d_out, int out_size, void* d_ws, size_t ws_size,
                              hipStream_t stream) {
  const float* p = (const float*)d_in[0];   // (1,T,1) f32
  const float* x = (const float*)d_in[1];   // (1,T,D) f32
  float* out = (float*)d_out;               // (D,) f32

  const int T = in_sizes[0];                // 1024
  const int D = in_sizes[1] / T;            // 256

  // workspace layout: c[T] | Q[T*T] | V[T*T]  (~8.4 MB)
  float* c = (float*)d_ws;
  float* Q = c + T;
  float* V = Q + (size_t)T * T;

  // two independent serial scans, one wave each, on separate blocks
  maa_scan_kernel<<<2, 32, 0, stream>>>(p, Q, V, T);
  // T parallel dot products (one wave per t)
  maa_dots_kernel<<<(T * 32 + 255) / 256, 256, 0, stream>>>(Q, V, p, c, T);
  // out = X^T c with f32 WMMA
  maa_matvec_wmma<<<D / 16, 32, 0, stream>>>(x, c, out, T, D);
}